// GraphConv_16604343566550
// MI455X (gfx1250) — compile-verified
//
#include <hip/hip_runtime.h>
#include <hip/hip_bf16.h>

typedef __attribute__((ext_vector_type(2))) float v2f;
typedef __attribute__((ext_vector_type(8))) float v8f;

#define D 128
#define LDS_STRIDE 132   // 132 mod 64 = 4 -> ds_load_b64 hits all 64 banks exactly once

// ---------------------------------------------------------------------------
// Kernel 0: zero the aggregation workspace (float4 stores, grid-stride)
// ---------------------------------------------------------------------------
__global__ void zero_ws_kernel(float4* __restrict__ ws, int n4) {
    int i = blockIdx.x * blockDim.x + threadIdx.x;
    int stride = gridDim.x * blockDim.x;
    float4 z = make_float4(0.f, 0.f, 0.f, 0.f);
    for (; i < n4; i += stride) ws[i] = z;
}

// ---------------------------------------------------------------------------
// Kernel 1: edge scatter. One wave32 per edge: lane L loads float4 chunk
// x[src][4L..4L+3], scales by edge_attr[e], atomically adds into agg[dst].
// Relaxed agent-scope fp32 atomics -> native global_atomic_add_f32 in L2.
// ---------------------------------------------------------------------------
__global__ void edge_scatter_kernel(const float* __restrict__ x,
                                    const int* __restrict__ edge_index, // [2,E] int32
                                    const float* __restrict__ edge_attr,
                                    float* __restrict__ agg,
                                    int E) {
    int gid  = blockIdx.x * blockDim.x + threadIdx.x;
    int e    = gid >> 5;        // wave id == edge id
    int lane = gid & 31;
    if (e >= E) return;

    int src = edge_index[e];
    int dst = edge_index[E + e];
    float w = edge_attr[e];

    const float4* xs = (const float4*)(x + (size_t)src * D);
    float4 v = xs[lane];                       // global_load_b128

    float* ap = agg + (size_t)dst * D + lane * 4;
    __hip_atomic_fetch_add(ap + 0, v.x * w, __ATOMIC_RELAXED, __HIP_MEMORY_SCOPE_AGENT);
    __hip_atomic_fetch_add(ap + 1, v.y * w, __ATOMIC_RELAXED, __HIP_MEMORY_SCOPE_AGENT);
    __hip_atomic_fetch_add(ap + 2, v.z * w, __ATOMIC_RELAXED, __HIP_MEMORY_SCOPE_AGENT);
    __hip_atomic_fetch_add(ap + 3, v.w * w, __ATOMIC_RELAXED, __HIP_MEMORY_SCOPE_AGENT);
}

// ---------------------------------------------------------------------------
// Kernel 2: fused out = agg @ W_rel^T + x @ W_root^T via V_WMMA_F32_16X16X4_F32.
//
// Per block: 16 node rows, 256 threads = 8 waves; wave ct owns output columns
// [16ct, 16ct+16). WMMA computes D = A x B + C with
//   A[m][k]  = W[16ct+m][kb+k]        (weights, per-lane b64 global loads)
//   B[k][n]  = V[row0+n][kb+k]        (node tile, b64 loads from padded LDS)
// so D[m][n] = out[row0+n][16ct+m] (transposed tile). f32 C/D layout then puts
// 8 *consecutive* output floats in each lane -> two float4 stores per lane.
// A-frag per lane: a[i] = A[lane&15][2*(lane>>4)+i]; B-frag symmetric.
// ---------------------------------------------------------------------------
__global__ __launch_bounds__(256) void graphconv_gemm_kernel(
        const float* __restrict__ x,
        const float* __restrict__ agg,
        const float* __restrict__ W_rel,
        const float* __restrict__ W_root,
        float* __restrict__ out,
        int N) {
    __shared__ float sAgg[16 * LDS_STRIDE];
    __shared__ float sX  [16 * LDS_STRIDE];

    const int row0 = blockIdx.x * 16;

    // Stage both 16x128 tiles into LDS (512 float4 per tile, 2 iters/thread).
    for (int i = threadIdx.x; i < 512; i += 256) {
        int r = i >> 5;            // 32 float4 per row
        int c = (i & 31) * 4;
        int row = row0 + r;
        if (row >= N) row = N - 1; // clamp (keeps loads in-bounds, no divergence)
        float4 va = *(const float4*)(agg + (size_t)row * D + c);
        float4 vx = *(const float4*)(x   + (size_t)row * D + c);
        *(float4*)(sAgg + r * LDS_STRIDE + c) = va;
        *(float4*)(sX   + r * LDS_STRIDE + c) = vx;
    }
    __syncthreads();

    const int waveId = threadIdx.x >> 5;   // column tile 0..7
    const int lane   = threadIdx.x & 31;
    const int m      = lane & 15;
    const int khi    = lane >> 4;          // 0 or 1

    const float* wr = W_rel  + (size_t)(waveId * 16 + m) * D; // A row (W_rel)
    const float* wo = W_root + (size_t)(waveId * 16 + m) * D; // A row (W_root)
    const float* vA = sAgg + m * LDS_STRIDE;                  // B row (agg tile)
    const float* vX = sX   + m * LDS_STRIDE;                  // B row (x tile)

    v8f c = {};

    #pragma unroll
    for (int kb = 0; kb < D; kb += 4) {
        int k = kb + 2 * khi;
        v2f a = *(const v2f*)(wr + k);
        v2f b = *(const v2f*)(vA + k);
        c = __builtin_amdgcn_wmma_f32_16x16x4_f32(false, a, false, b,
                                                  (short)0, c, false, false);
    }
    #pragma unroll
    for (int kb = 0; kb < D; kb += 4) {
        int k = kb + 2 * khi;
        v2f a = *(const v2f*)(wo + k);
        v2f b = *(const v2f*)(vX + k);
        c = __builtin_amdgcn_wmma_f32_16x16x4_f32(false, a, false, b,
                                                  (short)0, c, false, false);
    }

    // Lane holds out[row0+m][16*waveId + 8*khi + r], r = 0..7 (contiguous).
    int node = row0 + m;
    if (node < N) {
        float* op = out + (size_t)node * D + waveId * 16 + khi * 8;
        *(float4*)(op + 0) = make_float4(c[0], c[1], c[2], c[3]);
        *(float4*)(op + 4) = make_float4(c[4], c[5], c[6], c[7]);
    }
}

// ---------------------------------------------------------------------------
extern "C" void kernel_launch(void* const* d_in, const int* in_sizes, int n_in,
                              void* d_out, int out_size, void* d_ws, size_t ws_size,
                              hipStream_t stream) {
    const float* x          = (const float*)d_in[0];
    const int*   edge_index = (const int*)  d_in[1];
    const float* edge_attr  = (const float*)d_in[2];
    const float* W_rel      = (const float*)d_in[3];
    const float* W_root     = (const float*)d_in[4];
    float*       out        = (float*)d_out;
    float*       agg        = (float*)d_ws;      // N*D floats of scratch

    const int N = in_sizes[0] / D;               // 50000
    const int E = in_sizes[2];                   // 600000

    // 1) agg = 0
    int n4 = (N * D) / 4;
    int zgrid = (n4 + 255) / 256;
    if (zgrid > 4096) zgrid = 4096;
    zero_ws_kernel<<<zgrid, 256, 0, stream>>>((float4*)agg, n4);

    // 2) agg[dst] += w_e * x[src]   (wave per edge)
    long long threads = (long long)E * 32;
    int sgrid = (int)((threads + 255) / 256);
    edge_scatter_kernel<<<sgrid, 256, 0, stream>>>(x, edge_index, edge_attr, agg, E);

    // 3) out = agg @ W_rel^T + x @ W_root^T   (WMMA f32)
    int ggrid = (N + 15) / 16;
    graphconv_gemm_kernel<<<ggrid, 256, 0, stream>>>(x, agg, W_rel, W_root, out, N);
}